// RecurrentGCN_80599356277029
// MI455X (gfx1250) — compile-verified
//
#include <hip/hip_runtime.h>
#include <math.h>

#define N_NODES 100000
#define N_EDGES 1600000
#define IN_CH 128
#define HID 64
#define FULL_BLOCKS (N_NODES / 64)  // blocks [0, FULL_BLOCKS) have all 64 rows valid

typedef __attribute__((ext_vector_type(2))) float v2f;
typedef __attribute__((ext_vector_type(8))) float v8f;

__device__ __forceinline__ v8f wmma_f32(v2f a, v2f b, v8f c) {
  // D = A(16x4 f32) x B(4x16 f32) + C(16x16 f32), full fp32 precision
  return __builtin_amdgcn_wmma_f32_16x16x4_f32(false, a, false, b, (short)0, c,
                                               false, false);
}

__device__ __forceinline__ float sigmoid_f(float x) {
  return 1.0f / (1.0f + __expf(-x));
}

// ---------------- degree / norm kernels ----------------

__global__ void k_deg_init(float* deg) {
  unsigned i = blockIdx.x * blockDim.x + threadIdx.x;
  if (i < N_NODES) deg[i] = 1.0f;  // self-loop contribution
}

__global__ void k_deg_edges(const int* __restrict__ dst, float* deg) {
  unsigned e = blockIdx.x * blockDim.x + threadIdx.x;
  if (e < N_EDGES) atomicAdd(&deg[dst[e]], 1.0f);
}

__global__ void k_dinv(float* deg) {
  unsigned i = blockIdx.x * blockDim.x + threadIdx.x;
  if (i < N_NODES) deg[i] = rsqrtf(deg[i]);  // deg >= 1 always
}

__global__ void k_edge_norm(const int* __restrict__ src,
                            const int* __restrict__ dst,
                            const float* __restrict__ dinv,
                            float* __restrict__ norm) {
  unsigned e = blockIdx.x * blockDim.x + threadIdx.x;
  if (e < N_EDGES) norm[e] = dinv[src[e]] * dinv[dst[e]];
}

// ---------------- xw = x @ W  (100000x128 @ 128x64) ----------------
// One wave -> 16 rows x 64 cols. Block = 4 waves -> 64 rows.
// A rows clamped (branchless). The full/partial split is keyed on blockIdx
// (SGPR) -> pure scalar branch, unguarded clause stores on the fast path.

__global__ void k_xw(const float* __restrict__ x, const float* __restrict__ W,
                     float* __restrict__ xw) {
  const int wave = threadIdx.x >> 5;
  const int lane = threadIdx.x & 31;
  const int half = lane >> 4;
  const int lr = lane & 15;
  const int m0 = blockIdx.x * 64 + wave * 16;
  const int row = min(m0 + lr, N_NODES - 1);
  const float* xrow = x + (size_t)row * IN_CH;

  v8f c[4] = {};
  for (int kb = 0; kb < IN_CH / 4; ++kb) {
    const int k = kb * 4 + half * 2;
    v2f a = *(const v2f*)(xrow + k);
    const float* w0 = W + (size_t)k * HID;
#pragma unroll
    for (int t = 0; t < 4; ++t) {
      v2f b;
      b.x = w0[t * 16 + lr];
      b.y = w0[HID + t * 16 + lr];
      c[t] = wmma_f32(a, b, c[t]);
    }
  }

  float* outp = xw + (size_t)(m0 + half * 8) * HID + lr;
  if (blockIdx.x < FULL_BLOCKS) {  // scalar-uniform fast path
#pragma unroll
    for (int t = 0; t < 4; ++t)
#pragma unroll
      for (int j = 0; j < 8; ++j) outp[(size_t)j * HID + t * 16] = c[t][j];
  } else {
#pragma unroll
    for (int t = 0; t < 4; ++t)
#pragma unroll
      for (int j = 0; j < 8; ++j)
        if (m0 + half * 8 + j < N_NODES) outp[(size_t)j * HID + t * 16] = c[t][j];
  }
}

// ---------------- message aggregation ----------------

__global__ void k_selfloop(const float* __restrict__ dinv,
                           const float* __restrict__ xw,
                           float* __restrict__ agg) {
  unsigned idx = blockIdx.x * blockDim.x + threadIdx.x;
  if (idx < N_NODES * HID) {
    unsigned i = idx >> 6;
    float d = dinv[i];
    agg[idx] = d * d * xw[idx];
  }
}

__global__ void k_scatter(const int* __restrict__ src,
                          const int* __restrict__ dst,
                          const float* __restrict__ norm,
                          const float* __restrict__ xw,
                          float* __restrict__ agg) {
  unsigned idx = blockIdx.x * blockDim.x + threadIdx.x;
  if (idx < (unsigned)N_EDGES * HID) {
    unsigned e = idx >> 6;
    unsigned ch = idx & 63;
    int s = src[e];
    int d = dst[e];
    atomicAdd(&agg[(size_t)d * HID + ch], norm[e] * xw[(size_t)s * HID + ch]);
  }
}

__global__ void k_bias_relu(float* __restrict__ agg,
                            const float* __restrict__ b) {
  unsigned idx = blockIdx.x * blockDim.x + threadIdx.x;
  if (idx < N_NODES * HID) {
    float v = agg[idx] + b[idx & 63];
    agg[idx] = v > 0.0f ? v : 0.0f;
  }
}

// ---------------- fused GRU cell ----------------
// gi = hc @ w_ih.T, gh = hp @ w_hh.T ; r,z gates accumulate both GEMMs into
// shared accumulators (8 tiles, cols 0..127); i_n / h_n kept split (4 tiles
// each) for n = tanh(i_n + r*h_n). One wave: 16 rows x all 192 gate cols.

__global__ void k_gru(const float* __restrict__ hc,
                      const float* __restrict__ hp,
                      const float* __restrict__ w_ih,
                      const float* __restrict__ w_hh,
                      const float* __restrict__ b_ih,
                      const float* __restrict__ b_hh,
                      float* __restrict__ hn_out) {
  const int wave = threadIdx.x >> 5;
  const int lane = threadIdx.x & 31;
  const int half = lane >> 4;
  const int lr = lane & 15;
  const int m0 = blockIdx.x * 64 + wave * 16;
  const int row = min(m0 + lr, N_NODES - 1);  // branchless A-row clamp
  const float* hcrow = hc + (size_t)row * HID;
  const float* hprow = hp + (size_t)row * HID;

  v8f rz[8] = {};
  v8f an[4] = {};
  v8f hn[4] = {};

  for (int kb = 0; kb < HID / 4; ++kb) {
    const int k = kb * 4 + half * 2;
    v2f ax = *(const v2f*)(hcrow + k);
    v2f ah = *(const v2f*)(hprow + k);
#pragma unroll
    for (int t = 0; t < 8; ++t) {
      const int n = t * 16 + lr;  // w_ih.T[k, n] = w_ih[n*HID + k]
      v2f bi = *(const v2f*)(w_ih + (size_t)n * HID + k);
      rz[t] = wmma_f32(ax, bi, rz[t]);
      v2f bh = *(const v2f*)(w_hh + (size_t)n * HID + k);
      rz[t] = wmma_f32(ah, bh, rz[t]);
    }
#pragma unroll
    for (int t = 0; t < 4; ++t) {
      const int n = 2 * HID + t * 16 + lr;
      v2f bi = *(const v2f*)(w_ih + (size_t)n * HID + k);
      an[t] = wmma_f32(ax, bi, an[t]);
      v2f bh = *(const v2f*)(w_hh + (size_t)n * HID + k);
      hn[t] = wmma_f32(ah, bh, hn[t]);
    }
  }

  const int rbase = m0 + half * 8;
  const float* hpp = hp + (size_t)rbase * HID + lr;
  float* outp = hn_out + (size_t)rbase * HID + lr;

  if (blockIdx.x < FULL_BLOCKS) {  // scalar-uniform fast path
#pragma unroll
    for (int t = 0; t < 4; ++t) {
      const int col = t * 16 + lr;
      const float br = b_ih[col] + b_hh[col];
      const float bz = b_ih[HID + col] + b_hh[HID + col];
      const float bin = b_ih[2 * HID + col];
      const float bhn = b_hh[2 * HID + col];
#pragma unroll
      for (int j = 0; j < 8; ++j) {
        float rg = sigmoid_f(rz[t][j] + br);
        float zg = sigmoid_f(rz[t + 4][j] + bz);
        float ng = tanhf(an[t][j] + bin + rg * (hn[t][j] + bhn));
        float hpv = hpp[(size_t)j * HID + t * 16];
        outp[(size_t)j * HID + t * 16] = (1.0f - zg) * ng + zg * hpv;
      }
    }
  } else {
#pragma unroll
    for (int t = 0; t < 4; ++t) {
      const int col = t * 16 + lr;
      const float br = b_ih[col] + b_hh[col];
      const float bz = b_ih[HID + col] + b_hh[HID + col];
      const float bin = b_ih[2 * HID + col];
      const float bhn = b_hh[2 * HID + col];
#pragma unroll
      for (int j = 0; j < 8; ++j) {
        if (rbase + j < N_NODES) {
          float rg = sigmoid_f(rz[t][j] + br);
          float zg = sigmoid_f(rz[t + 4][j] + bz);
          float ng = tanhf(an[t][j] + bin + rg * (hn[t][j] + bhn));
          float hpv = hpp[(size_t)j * HID + t * 16];
          outp[(size_t)j * HID + t * 16] = (1.0f - zg) * ng + zg * hpv;
        }
      }
    }
  }
}

// ---------------- fused edge MLP ----------------
// hdn = relu([h[src]|h[dst]|ea] @ ec_w1 + b1); out = hdn @ ec_w2 + b2
// One wave: 16 edges x 64 hidden. Block 256 = 8 waves = 128 edges.
// N_EDGES % 128 == 0 so no edge guards needed.

__global__ void k_edge_mlp(const int* __restrict__ src,
                           const int* __restrict__ dst,
                           const float* __restrict__ hnext,
                           const float* __restrict__ eattr,
                           const float* __restrict__ ec_w1,
                           const float* __restrict__ ec_b1,
                           const float* __restrict__ ec_w2,
                           const float* __restrict__ ec_b2,
                           float* __restrict__ out) {
  const int wave = threadIdx.x >> 5;
  const int lane = threadIdx.x & 31;
  const int half = lane >> 4;
  const int lr = lane & 15;
  const int e0 = blockIdx.x * 128 + wave * 16;

  const size_t sbase = (size_t)src[e0 + lr] * HID;
  const size_t dbase = (size_t)dst[e0 + lr] * HID;

  v8f c[4] = {};
  // src half: K = 0..63 of ec_w1
  for (int kb = 0; kb < HID / 4; ++kb) {
    const int k = kb * 4 + half * 2;
    v2f a = *(const v2f*)(hnext + sbase + k);
    const float* w0 = ec_w1 + (size_t)k * HID;
#pragma unroll
    for (int t = 0; t < 4; ++t) {
      v2f b;
      b.x = w0[t * 16 + lr];
      b.y = w0[HID + t * 16 + lr];
      c[t] = wmma_f32(a, b, c[t]);
    }
  }
  // dst half: K = 64..127 of ec_w1
  for (int kb = 0; kb < HID / 4; ++kb) {
    const int k = kb * 4 + half * 2;
    v2f a = *(const v2f*)(hnext + dbase + k);
    const float* w0 = ec_w1 + (size_t)(HID + k) * HID;
#pragma unroll
    for (int t = 0; t < 4; ++t) {
      v2f b;
      b.x = w0[t * 16 + lr];
      b.y = w0[HID + t * 16 + lr];
      c[t] = wmma_f32(a, b, c[t]);
    }
  }

  float w1c0[4], w1c1[4], b1v[4], w2v[4];
#pragma unroll
  for (int t = 0; t < 4; ++t) {
    const int col = t * 16 + lr;
    w1c0[t] = ec_w1[(size_t)(2 * HID) * HID + col];
    w1c1[t] = ec_w1[(size_t)(2 * HID + 1) * HID + col];
    b1v[t] = ec_b1[col];
    w2v[t] = ec_w2[col];
  }
  const float b2 = ec_b2[0];

#pragma unroll
  for (int j = 0; j < 8; ++j) {
    const int erow = e0 + half * 8 + j;
    const float ea0 = eattr[(size_t)erow * 2];
    const float ea1 = eattr[(size_t)erow * 2 + 1];
    float acc = 0.0f;
#pragma unroll
    for (int t = 0; t < 4; ++t) {
      float h = c[t][j] + ea0 * w1c0[t] + ea1 * w1c1[t] + b1v[t];
      h = fmaxf(h, 0.0f);
      acc += h * w2v[t];
    }
    // reduce over 16 lanes of this half-wave (cols)
    acc += __shfl_xor(acc, 1, 32);
    acc += __shfl_xor(acc, 2, 32);
    acc += __shfl_xor(acc, 4, 32);
    acc += __shfl_xor(acc, 8, 32);
    if (lr == 0) out[erow] = acc + b2;
  }
}

// ---------------- host launcher ----------------

static inline unsigned cdiv(unsigned a, unsigned b) { return (a + b - 1) / b; }

extern "C" void kernel_launch(void* const* d_in, const int* in_sizes, int n_in,
                              void* d_out, int out_size, void* d_ws,
                              size_t ws_size, hipStream_t stream) {
  const float* x = (const float*)d_in[0];
  const int* ei = (const int*)d_in[1];
  const float* eattr = (const float*)d_in[2];
  const float* h_prev = (const float*)d_in[3];
  const float* conv_w = (const float*)d_in[4];
  const float* conv_b = (const float*)d_in[5];
  const float* w_ih = (const float*)d_in[6];
  const float* w_hh = (const float*)d_in[7];
  const float* b_ih = (const float*)d_in[8];
  const float* b_hh = (const float*)d_in[9];
  const float* ec_w1 = (const float*)d_in[10];
  const float* ec_b1 = (const float*)d_in[11];
  const float* ec_w2 = (const float*)d_in[12];
  const float* ec_b2 = (const float*)d_in[13];

  const int* src = ei;
  const int* dst = ei + N_EDGES;

  float* out = (float*)d_out;    // (N_EDGES,)
  float* h_next = out + N_EDGES; // (N_NODES, HID)

  float* w = (float*)d_ws;
  float* deg = w;                             // N_NODES (deg -> dinv in place)
  float* xw = deg + N_NODES;                  // N_NODES*HID
  float* agg = xw + (size_t)N_NODES * HID;    // N_NODES*HID (agg -> h_curr)
  float* norm = agg + (size_t)N_NODES * HID;  // N_EDGES

  const unsigned elem_n = N_NODES * HID;
  const unsigned elem_e = (unsigned)N_EDGES * HID;
  const unsigned row_blocks = cdiv(N_NODES, 64);

  k_deg_init<<<cdiv(N_NODES, 256), 256, 0, stream>>>(deg);
  k_deg_edges<<<cdiv(N_EDGES, 256), 256, 0, stream>>>(dst, deg);
  k_dinv<<<cdiv(N_NODES, 256), 256, 0, stream>>>(deg);
  k_edge_norm<<<cdiv(N_EDGES, 256), 256, 0, stream>>>(src, dst, deg, norm);

  k_xw<<<row_blocks, 128, 0, stream>>>(x, conv_w, xw);

  k_selfloop<<<cdiv(elem_n, 256), 256, 0, stream>>>(deg, xw, agg);
  k_scatter<<<cdiv(elem_e, 256), 256, 0, stream>>>(src, dst, norm, xw, agg);
  k_bias_relu<<<cdiv(elem_n, 256), 256, 0, stream>>>(agg, conv_b);

  k_gru<<<row_blocks, 128, 0, stream>>>(agg, h_prev, w_ih, w_hh, b_ih, b_hh,
                                        h_next);

  k_edge_mlp<<<N_EDGES / 128, 256, 0, stream>>>(src, dst, h_next, eattr, ec_w1,
                                                ec_b1, ec_w2, ec_b2, out);
}